// HashGridEncoding_36782099923509
// MI455X (gfx1250) — compile-verified
//
#include <hip/hip_runtime.h>
#include <hip/hip_bf16.h>
#include <stdint.h>

// ---------------------------------------------------------------------------
// HashGrid encoding for MI455X (gfx1250, wave32).
// One thread = one point, all 12 levels unrolled. Coordinates are staged
// into LDS with gfx1250 async-to-LDS copies (ASYNCcnt), double-buffered
// across a grid-stride loop. Output rows (96 B) stored with NT hint so the
// 96 MB write stream does not evict the 48 MB of tables from the 192 MB L2.
// ---------------------------------------------------------------------------

#define NLVL   12
#define TSIZE  (1u << 19)
#define TMASK  (TSIZE - 1u)
#define NPTS   1048576
#define BLOCK  256
#define NCHUNK (NPTS / BLOCK)   // 4096 chunks of 256 points
#define GRID   1024             // 4 chunks per block -> prefetch overlap

typedef float v4f __attribute__((ext_vector_type(4)));

// gfx1250 async global->LDS copy (tracked with ASYNCcnt).
// VDST = per-lane LDS byte address (low 32 bits of the generic pointer),
// VADDR = per-lane 64-bit global address.
__device__ __forceinline__ void async_ld_b32(uint32_t lds_addr, const float* gaddr) {
    asm volatile("global_load_async_to_lds_b32 %0, %1, off"
                 :
                 : "v"(lds_addr), "v"(gaddr)
                 : "memory");
}

__device__ __forceinline__ void wait_async_zero() {
    asm volatile("s_wait_asynccnt 0" ::: "memory");
}

__device__ __forceinline__ void stage_chunk(float* sbuf, const float* gsrc, int tid) {
    // 256 threads copy 768 contiguous dwords (256 pts x 3 floats), coalesced:
    // each lane copies dwords tid, tid+256, tid+512.
    uint32_t l = (uint32_t)(uintptr_t)(sbuf + tid);
    const float* g = gsrc + tid;
    async_ld_b32(l,                 g);
    async_ld_b32(l + BLOCK * 4u,    g + BLOCK);
    async_ld_b32(l + 2u * BLOCK * 4u, g + 2 * BLOCK);
}

__global__ __launch_bounds__(BLOCK)
void HashGridEncoding_36782099923509_kernel(const float* __restrict__ x,
                                            const float* __restrict__ tables,
                                            float* __restrict__ out) {
    // floor(16 * 1.38^l) for l = 0..11, as exact f32 constants
    constexpr float RES[NLVL] = {16.f, 22.f, 30.f, 42.f, 58.f, 80.f,
                                 110.f, 152.f, 210.f, 290.f, 400.f, 553.f};
    constexpr uint32_t P1 = 2654435761u;
    constexpr uint32_t P2 = 805459861u;

    __shared__ float sx[2][BLOCK * 3];
    const int tid = threadIdx.x;

    // Prologue: stage this block's first chunk into buffer 0.
    stage_chunk(&sx[0][0], x + (size_t)blockIdx.x * (BLOCK * 3), tid);

    int buf = 0;
    for (int chunk = blockIdx.x; chunk < NCHUNK; chunk += gridDim.x) {
        // Drain this wave's async copies, then barrier so every wave's
        // copies into sx[buf] are visible (and everyone is done reading
        // sx[buf^1] from the previous iteration before we overwrite it).
        wait_async_zero();
        __syncthreads();

        const float px = sx[buf][3 * tid + 0];
        const float py = sx[buf][3 * tid + 1];
        const float pz = sx[buf][3 * tid + 2];

        // Prefetch the next chunk into the other buffer; overlaps with the
        // 96 table gathers + interpolation below.
        const int nchunk = chunk + (int)gridDim.x;
        if (nchunk < NCHUNK)
            stage_chunk(&sx[buf ^ 1][0], x + (size_t)nchunk * (BLOCK * 3), tid);

        float acc[2 * NLVL];
#pragma unroll
        for (int l = 0; l < NLVL; ++l) {
            const float res = RES[l];
            const float xs = px * res, ys = py * res, zs = pz * res;
            const float fx = floorf(xs), fy = floorf(ys), fz = floorf(zs);
            const float wx = xs - fx, wy = ys - fy, wz = zs - fz;
            const uint32_t xi = (uint32_t)(int)fx;
            const uint32_t yi = (uint32_t)(int)fy;
            const uint32_t zi = (uint32_t)(int)fz;

            // h = cx ^ (cy*P1) ^ (cz*P2), uint32 wraparound, idx = h & (T-1)
            const uint32_t hy0 = yi * P1, hy1 = hy0 + P1;
            const uint32_t hz0 = zi * P2, hz1 = hz0 + P2;
            const uint32_t b00 = hy0 ^ hz0, b10 = hy1 ^ hz0;
            const uint32_t b01 = hy0 ^ hz1, b11 = hy1 ^ hz1;
            const uint32_t x0 = xi, x1 = xi + 1u;

            const uint32_t i000 = (x0 ^ b00) & TMASK;
            const uint32_t i100 = (x1 ^ b00) & TMASK;
            const uint32_t i010 = (x0 ^ b10) & TMASK;
            const uint32_t i110 = (x1 ^ b10) & TMASK;
            const uint32_t i001 = (x0 ^ b01) & TMASK;
            const uint32_t i101 = (x1 ^ b01) & TMASK;
            const uint32_t i011 = (x0 ^ b11) & TMASK;
            const uint32_t i111 = (x1 ^ b11) & TMASK;

            const float2* tbl = reinterpret_cast<const float2*>(tables) + (size_t)l * TSIZE;
            // 8 independent b64 gathers (L2-resident tables) -> one clause.
            const float2 f000 = tbl[i000];
            const float2 f100 = tbl[i100];
            const float2 f010 = tbl[i010];
            const float2 f110 = tbl[i110];
            const float2 f001 = tbl[i001];
            const float2 f101 = tbl[i101];
            const float2 f011 = tbl[i011];
            const float2 f111 = tbl[i111];

            const float ox = 1.f - wx, oy = 1.f - wy, oz = 1.f - wz;
            const float w000 = ox * oy * oz, w100 = wx * oy * oz;
            const float w010 = ox * wy * oz, w110 = wx * wy * oz;
            const float w001 = ox * oy * wz, w101 = wx * oy * wz;
            const float w011 = ox * wy * wz, w111 = wx * wy * wz;

            float a0 = w000 * f000.x;
            a0 = fmaf(w100, f100.x, a0);
            a0 = fmaf(w010, f010.x, a0);
            a0 = fmaf(w110, f110.x, a0);
            a0 = fmaf(w001, f001.x, a0);
            a0 = fmaf(w101, f101.x, a0);
            a0 = fmaf(w011, f011.x, a0);
            a0 = fmaf(w111, f111.x, a0);

            float a1 = w000 * f000.y;
            a1 = fmaf(w100, f100.y, a1);
            a1 = fmaf(w010, f010.y, a1);
            a1 = fmaf(w110, f110.y, a1);
            a1 = fmaf(w001, f001.y, a1);
            a1 = fmaf(w101, f101.y, a1);
            a1 = fmaf(w011, f011.y, a1);
            a1 = fmaf(w111, f111.y, a1);

            acc[2 * l]     = a0;
            acc[2 * l + 1] = a1;
        }

        // 96 B contiguous per point, NT hint: don't let the output stream
        // evict the L2-resident hash tables.
        const size_t gidx = (size_t)chunk * BLOCK + tid;
        float* op = out + gidx * (2 * NLVL);
#pragma unroll
        for (int k = 0; k < 6; ++k) {
            v4f v = {acc[4 * k + 0], acc[4 * k + 1], acc[4 * k + 2], acc[4 * k + 3]};
            __builtin_nontemporal_store(v, reinterpret_cast<v4f*>(op) + k);
        }

        buf ^= 1;
    }
}

extern "C" void kernel_launch(void* const* d_in, const int* in_sizes, int n_in,
                              void* d_out, int out_size, void* d_ws, size_t ws_size,
                              hipStream_t stream) {
    (void)in_sizes; (void)n_in; (void)out_size; (void)d_ws; (void)ws_size;
    const float* x      = (const float*)d_in[0];   // (N_PTS, 3) f32
    const float* tables = (const float*)d_in[1];   // (12, 2^19, 2) f32
    float* out          = (float*)d_out;           // (N_PTS, 24) f32

    HashGridEncoding_36782099923509_kernel<<<dim3(GRID), dim3(BLOCK), 0, stream>>>(
        x, tables, out);
}